// DPQNetwork_11510512353918
// MI455X (gfx1250) — compile-verified
//
#include <hip/hip_runtime.h>

// ---------------------------------------------------------------------------
// DPQ encode: response = einsum('bcd,ckd->bck'), argmax over k, gather centroid.
// B=131072, C=32, K=16, D=8.  HBM-bound (~272MB @ 23.3TB/s ~= 12us).
// Per-codebook GEMM [16x8]x[8x16] -> two chained V_WMMA_F32_16X16X4_F32.
// Argmax: WMMA tile -> wave-private LDS scratch -> branchless serial argmax
// (no ds_bpermute, no exec-mask branching).
// ---------------------------------------------------------------------------

typedef __attribute__((ext_vector_type(2))) float v2f;
typedef __attribute__((ext_vector_type(8))) float v8f;

#define B_TOT 131072
#define C_NB  32
#define K_NC  16
#define D_SV  8

#define ROWS       32   // batch rows staged per block
#define IN_STRIDE  260  // 256 floats/row + 4 pad  (260 % 64 == 4 -> conflict-free)
#define CT_STRIDE  12   // 8 floats/centroid + 4 pad (float4-aligned)
#define RED_STRIDE 20   // 16 floats/row + 4 pad (b128-aligned, banks tile 0..63)

__global__ __launch_bounds__(256)
void dpq_wmma_kernel(const float* __restrict__ inputs,
                     const float* __restrict__ centroids,
                     float* __restrict__ codes,
                     float* __restrict__ outs) {
    __shared__ __align__(16) float sIn[ROWS * IN_STRIDE];           // 33,280 B
    __shared__ __align__(16) float sCt[C_NB * K_NC * CT_STRIDE];    // 24,576 B
    __shared__ __align__(16) float sRed[8 * 16 * RED_STRIDE];       // 10,240 B

    const int tid   = threadIdx.x;
    const int bBase = blockIdx.x * ROWS;

    // ---- stage input tile, fully coalesced float4 loads (8 per thread) ----
    {
        const float4* gin = (const float4*)(inputs + (size_t)bBase * (C_NB * D_SV));
        #pragma unroll
        for (int i = 0; i < (ROWS * C_NB * D_SV) / 4 / 256; ++i) {
            int f4  = tid + i * 256;
            int row = f4 >> 6;            // 64 float4 per staged row
            int col = (f4 & 63) << 2;
            float4 v = gin[f4];
            *(float4*)&sIn[row * IN_STRIDE + col] = v;
        }
    }
    // ---- stage all centroids (16 KB), coalesced float4 (4 per thread) ----
    {
        const float4* gct = (const float4*)centroids;
        #pragma unroll
        for (int i = 0; i < (C_NB * K_NC * D_SV) / 4 / 256; ++i) {
            int f4  = tid + i * 256;
            int row = f4 >> 1;            // 2 float4 per centroid row
            int col = (f4 & 1) << 2;
            float4 v = gct[f4];
            *(float4*)&sCt[row * CT_STRIDE + col] = v;
        }
    }
    __syncthreads();

    const int wave = tid >> 5;
    const int lane = tid & 31;
    const int g    = lane >> 4;   // half-wave: selects K-dim pair / row offset +8
    const int m    = lane & 15;   // A: row M; B: col N

    float* wred = &sRed[wave * 16 * RED_STRIDE];   // wave-private scratch

    // 64 tasks per block: 2 row-subtiles x 32 codebooks; 8 per wave.
    #pragma unroll
    for (int t8 = 0; t8 < 8; ++t8) {
        const int task = wave * 8 + t8;
        const int c    = task & 31;
        const int sub  = task >> 5;

        // A frags: lane holds input row (sub*16+m), K-dims {d0,d0+1} per WMMA
        const float* arow = &sIn[(sub * 16 + m) * IN_STRIDE + c * D_SV];
        // B frags: lane holds centroid n=m of codebook c, same K-dim split
        const float* brow = &sCt[(c * K_NC + m) * CT_STRIDE];
        const int d0 = 2 * g;

        v2f fa0 = { arow[d0],     arow[d0 + 1]     };
        v2f fb0 = { brow[d0],     brow[d0 + 1]     };
        v2f fa1 = { arow[4 + d0], arow[4 + d0 + 1] };
        v2f fb1 = { brow[4 + d0], brow[4 + d0 + 1] };

        v8f acc = {0.f, 0.f, 0.f, 0.f, 0.f, 0.f, 0.f, 0.f};
        acc = __builtin_amdgcn_wmma_f32_16x16x4_f32(false, fa0, false, fb0,
                                                    (short)0, acc, false, false);
        acc = __builtin_amdgcn_wmma_f32_16x16x4_f32(false, fa1, false, fb1,
                                                    (short)0, acc, false, false);
        // acc[r] @ lane: response[M = r + 8*g][N = m]

        // ---- dump tile row-major into wave-private scratch (conflict-free) ----
        #pragma unroll
        for (int r = 0; r < 8; ++r)
            wred[(r + 8 * g) * RED_STRIDE + m] = acc[r];
        // same-wave LDS ops are in-order (DScnt); no barrier needed.

        // ---- lanes 0..15: branchless serial argmax over own row ----
        if (lane < 16) {
            const float* rowp = &wred[lane * RED_STRIDE];
            float4 q0 = *(const float4*)(rowp + 0);
            float4 q1 = *(const float4*)(rowp + 4);
            float4 q2 = *(const float4*)(rowp + 8);
            float4 q3 = *(const float4*)(rowp + 12);
            float vals[16] = { q0.x, q0.y, q0.z, q0.w,
                               q1.x, q1.y, q1.z, q1.w,
                               q2.x, q2.y, q2.z, q2.w,
                               q3.x, q3.y, q3.z, q3.w };
            float best = vals[0];
            int   bi   = 0;
            #pragma unroll
            for (int j = 1; j < 16; ++j) {
                bool t = vals[j] > best;        // strict > keeps first max (jnp.argmax)
                best = t ? vals[j] : best;
                bi   = t ? j      : bi;
            }

            const int  rowG = bBase + sub * 16 + lane;
            const long base = (long)rowG * C_NB + c;
            codes[base] = (float)(bi + c * K_NC);       // offset code

            // gather winning centroid row (8 floats) -> two b128 stores
            const float4* cp = (const float4*)&sCt[(c * K_NC + bi) * CT_STRIDE];
            float4 o0 = cp[0];
            float4 o1 = cp[1];
            float4* op = (float4*)&outs[base * D_SV];
            op[0] = o0;
            op[1] = o1;
        }
    }
}

extern "C" void kernel_launch(void* const* d_in, const int* in_sizes, int n_in,
                              void* d_out, int out_size, void* d_ws, size_t ws_size,
                              hipStream_t stream) {
    const float* inputs    = (const float*)d_in[0];   // [B, C, D] f32
    const float* centroids = (const float*)d_in[1];   // [C, K, D] f32
    float* codes = (float*)d_out;                     // [B, C]  (codes cast to f32)
    float* outs  = codes + (size_t)B_TOT * C_NB;      // [B, C, D]

    dim3 grid(B_TOT / ROWS), block(256);
    hipLaunchKernelGGL(dpq_wmma_kernel, grid, block, 0, stream,
                       inputs, centroids, codes, outs);
}